// EdgeGatedGraphConv_30666066493673
// MI455X (gfx1250) — compile-verified
//
#include <hip/hip_runtime.h>
#include <hip/hip_bf16.h>
#include <cstddef>

// Problem constants (fixed by the reference)
#define BB 4
#define NN 4096
#define EE 65536
#define FF 128
#define LDSPAD 132   // padded LDS row stride (floats) -> bank-conflict-free column reads

typedef __attribute__((ext_vector_type(2))) float v2f;
typedef __attribute__((ext_vector_type(8))) float v8f;
typedef __attribute__((ext_vector_type(4))) int v4i;

#if defined(__has_builtin)
#if __has_builtin(__builtin_amdgcn_wmma_f32_16x16x4_f32)
#define HAS_F32_WMMA 1
#endif
#if __has_builtin(__builtin_amdgcn_global_load_async_to_lds_b128) && \
    __has_builtin(__builtin_amdgcn_s_wait_asynccnt)
#define HAS_ASYNC_LDS 1
#endif
#endif

// Pointer types for the async global->LDS DMA builtin:
// arg0: v4i in address_space(1) (prints as "__device__"), arg1: v4i in AS(3).
typedef __attribute__((address_space(1))) v4i g_v4i;
typedef __attribute__((address_space(3))) v4i l_v4i;

// Stage a contiguous 16x128 fp32 tile (8 KB) into LDS with padded row stride.
// Prefers gfx1250 async global->LDS DMA (no VGPR round trip, ASYNCcnt-tracked).
static __device__ __forceinline__ void stage_tile_16x128(
    float* lx, const float* gsrc, int tid) {
#if defined(HAS_ASYNC_LDS)
  for (int t = tid; t < (16 * FF) / 4; t += 256) {
    int r = (t * 4) / FF, c = (t * 4) % FF;
    __builtin_amdgcn_global_load_async_to_lds_b128(
        (g_v4i*)(gsrc + t * 4), (l_v4i*)&lx[r * LDSPAD + c],
        /*offset=*/0, /*cpol=*/0);
  }
  __builtin_amdgcn_s_wait_asynccnt(0);
#else
  const float4* src = (const float4*)gsrc;
  for (int t = tid; t < (16 * FF) / 4; t += 256) {
    float4 v = src[t];
    int r = (t * 4) / FF, c = (t * 4) % FF;
    *(float4*)&lx[r * LDSPAD + c] = v;
  }
#endif
}

// ---------------------------------------------------------------------------
// Kernel 0: zero the scatter accumulators (sum_sigma_h, sum_sigma)
// ---------------------------------------------------------------------------
__global__ void zero_kernel(float* __restrict__ p, int n) {
  int idx = (blockIdx.x * blockDim.x + threadIdx.x) * 4;
  if (idx < n) *(float4*)(p + idx) = make_float4(0.f, 0.f, 0.f, 0.f);
}

// ---------------------------------------------------------------------------
// Kernel 1: four node linears  O_m = X @ W_m^T + b_m   (X: [B*N, F])
// Block: 256 threads (8 wave32). Each block owns a 16-row tile of X (staged
// in LDS); wave w owns output columns [16w, 16w+16) across all 4 matrices.
// Per 16x16 output tile: 32x v_wmma_f32_16x16x4_f32 over K=128.
// ---------------------------------------------------------------------------
__global__ __launch_bounds__(256)
void node_linear4_kernel(const float* __restrict__ X,
                         const float* __restrict__ W0, const float* __restrict__ bv0,
                         const float* __restrict__ W1, const float* __restrict__ bv1,
                         const float* __restrict__ W2, const float* __restrict__ bv2,
                         const float* __restrict__ W3, const float* __restrict__ bv3,
                         float* __restrict__ O0, float* __restrict__ O1,
                         float* __restrict__ O2, float* __restrict__ O3) {
  __shared__ float lx[16 * LDSPAD];
  const int tid = threadIdx.x;
  const int rowbase = blockIdx.x * 16;

  stage_tile_16x128(lx, X + (size_t)rowbase * FF, tid);
  __syncthreads();

  const int wid = tid >> 5;
  const int lane = tid & 31;
  const int nlane = lane & 15;
  const int hi = lane >> 4;
  const int gbase = wid * 16;

  // A fragments (ISA 32-bit 16x4 layout): lane holds row (lane&15),
  // K = t*4 + 2*(lane>>4) + {0,1}  -> one ds_load_b64 per K-step.
  v2f afrag[32];
#pragma unroll
  for (int t = 0; t < 32; ++t)
    afrag[t] = *(const v2f*)&lx[nlane * LDSPAD + t * 4 + hi * 2];

  const float* Ws[4] = {W0, W1, W2, W3};
  const float* bs[4] = {bv0, bv1, bv2, bv3};
  float* Os[4] = {O0, O1, O2, O3};

#pragma unroll
  for (int m = 0; m < 4; ++m) {
    const float* Wrow = Ws[m] + (size_t)(gbase + nlane) * FF;  // B[k][n] = W[n][k]
    float bias = bs[m][gbase + nlane];
    v8f c;
#pragma unroll
    for (int r = 0; r < 8; ++r) c[r] = bias;
#if defined(HAS_F32_WMMA)
#pragma unroll
    for (int t = 0; t < 32; ++t) {
      v2f bf = *(const v2f*)(Wrow + t * 4 + hi * 2);
      c = __builtin_amdgcn_wmma_f32_16x16x4_f32(false, afrag[t], false, bf,
                                                (short)0, c, false, false);
    }
#else
    for (int r = 0; r < 8; ++r) {
      int mr = r + 8 * hi;
      float acc = c[r];
      for (int k = 0; k < FF; ++k) acc += lx[mr * LDSPAD + k] * Wrow[k];
      c[r] = acc;
    }
#endif
    // D layout: VGPR r, lane l -> (row = r + 8*(l>>4), col = gbase + (l&15))
    float* orow = Os[m] + (size_t)rowbase * FF + gbase + nlane;
#pragma unroll
    for (int r = 0; r < 8; ++r) orow[(size_t)(r + 8 * hi) * FF] = c[r];
  }
}

// ---------------------------------------------------------------------------
// Kernel 2: edge pipeline. Per block: 16 edges x 128 cols.
//   eg = edge_feats @ W_edge_gate^T + b  (WMMA, same micro-kernel)
//   y  = src_gate[i] + dst_gate[j] + eg ; sig = sigmoid(y)
//   y_out = edge_feats + y*sig ; atomic scatter of (dst_update[j]*sig, sig)
// ---------------------------------------------------------------------------
__global__ __launch_bounds__(256)
void edge_kernel(const float* __restrict__ edge_feats,
                 const int* __restrict__ iidx, const int* __restrict__ jidx,
                 const float* __restrict__ Wg, const float* __restrict__ bg,
                 const float* __restrict__ src_gate,
                 const float* __restrict__ dst_gate,
                 const float* __restrict__ dst_update,
                 float* __restrict__ y_out,
                 float* __restrict__ ssh, float* __restrict__ ss) {
  __shared__ float lx[16 * LDSPAD];
  __shared__ int li[16], lj[16];
  const int tid = threadIdx.x;
  const int b = blockIdx.x / (EE / 16);
  const int e0 = (blockIdx.x % (EE / 16)) * 16;

  // Index loads + L2 prefetch of gather targets overlap the async tile DMA.
  if (tid < 16) {
    int v = iidx[e0 + tid];
    li[tid] = v;
    __builtin_prefetch(&src_gate[(size_t)(b * NN + v) * FF], 0, 1);  // global_prefetch_b8
    __builtin_prefetch(&ssh[(size_t)(b * NN + v) * FF], 1, 1);
  } else if (tid < 32) {
    int v = jidx[e0 + tid - 16];
    lj[tid - 16] = v;
    __builtin_prefetch(&dst_gate[(size_t)(b * NN + v) * FF], 0, 1);
    __builtin_prefetch(&dst_update[(size_t)(b * NN + v) * FF], 0, 1);
  }
  stage_tile_16x128(lx, edge_feats + (size_t)(b * EE + e0) * FF, tid);
  __syncthreads();

  const int lane = tid & 31;
  const int nlane = lane & 15;
  const int hi = lane >> 4;
  const int gbase = (tid >> 5) * 16;

  v2f afrag[32];
#pragma unroll
  for (int t = 0; t < 32; ++t)
    afrag[t] = *(const v2f*)&lx[nlane * LDSPAD + t * 4 + hi * 2];

  const float* Wrow = Wg + (size_t)(gbase + nlane) * FF;
  float bias = bg[gbase + nlane];
  v8f c;
#pragma unroll
  for (int r = 0; r < 8; ++r) c[r] = bias;
#if defined(HAS_F32_WMMA)
#pragma unroll
  for (int t = 0; t < 32; ++t) {
    v2f bf = *(const v2f*)(Wrow + t * 4 + hi * 2);
    c = __builtin_amdgcn_wmma_f32_16x16x4_f32(false, afrag[t], false, bf,
                                              (short)0, c, false, false);
  }
#else
  for (int r = 0; r < 8; ++r) {
    int mr = r + 8 * hi;
    float acc = c[r];
    for (int k = 0; k < FF; ++k) acc += lx[mr * LDSPAD + k] * Wrow[k];
    c[r] = acc;
  }
#endif

  const int col = gbase + nlane;
#pragma unroll
  for (int r = 0; r < 8; ++r) {
    int m = r + 8 * hi;  // edge row within the tile
    size_t nI = (size_t)(b * NN + li[m]) * FF + col;
    size_t nJ = (size_t)(b * NN + lj[m]) * FF + col;
    float es = src_gate[nI];
    float ed = dst_gate[nJ];
    float bh = dst_update[nJ];
    float y = es + ed + c[r];
    float sig = 1.0f / (1.0f + __expf(-y));
    float ef = lx[m * LDSPAD + col];
    y_out[(size_t)(b * EE + e0 + m) * FF + col] = ef + y * sig;  // silu(y)=y*sig
    __hip_atomic_fetch_add(&ssh[nI], bh * sig, __ATOMIC_RELAXED,
                           __HIP_MEMORY_SCOPE_AGENT);
    __hip_atomic_fetch_add(&ss[nI], sig, __ATOMIC_RELAXED,
                           __HIP_MEMORY_SCOPE_AGENT);
  }
}

// ---------------------------------------------------------------------------
// Kernel 3: x = node + silu(src_update + ssh/(ss+1e-6))
// ---------------------------------------------------------------------------
__global__ void finalize_kernel(const float* __restrict__ node,
                                const float* __restrict__ su,
                                const float* __restrict__ ssh,
                                const float* __restrict__ ss,
                                float* __restrict__ x_out, int n) {
  int idx = blockIdx.x * blockDim.x + threadIdx.x;
  if (idx < n) {
    float h = ssh[idx] / (ss[idx] + 1e-6f);
    float t = su[idx] + h;
    x_out[idx] = node[idx] + t / (1.0f + __expf(-t));
  }
}

// ---------------------------------------------------------------------------
extern "C" void kernel_launch(void* const* d_in, const int* in_sizes, int n_in,
                              void* d_out, int out_size, void* d_ws,
                              size_t ws_size, hipStream_t stream) {
  (void)in_sizes; (void)n_in; (void)out_size; (void)ws_size;
  const float* node = (const float*)d_in[0];
  const float* edge = (const float*)d_in[1];
  const int* ii = (const int*)d_in[2];
  const int* jj = (const int*)d_in[3];
  const float* Wsg = (const float*)d_in[4];
  const float* bsg = (const float*)d_in[5];
  const float* Wdg = (const float*)d_in[6];
  const float* bdg = (const float*)d_in[7];
  const float* Weg = (const float*)d_in[8];
  const float* beg = (const float*)d_in[9];
  const float* Wsu = (const float*)d_in[10];
  const float* bsu = (const float*)d_in[11];
  const float* Wdu = (const float*)d_in[12];
  const float* bdu = (const float*)d_in[13];

  const size_t NNF = (size_t)BB * NN * FF;  // 2,097,152 floats (8 MB)
  float* ws = (float*)d_ws;                 // need 6*NNF*4 = 48 MB
  float* src_gate = ws;
  float* dst_gate = ws + NNF;
  float* dst_update = ws + 2 * NNF;
  float* src_update = ws + 3 * NNF;
  float* ssh = ws + 4 * NNF;
  float* ssum = ws + 5 * NNF;

  float* x_out = (float*)d_out;  // [B,N,F]
  float* y_out = x_out + NNF;    // [B,E,F]

  zero_kernel<<<(int)(2 * NNF / 4 / 256), 256, 0, stream>>>(ssh, (int)(2 * NNF));

  node_linear4_kernel<<<(BB * NN) / 16, 256, 0, stream>>>(
      node, Wsg, bsg, Wdg, bdg, Wdu, bdu, Wsu, bsu,
      src_gate, dst_gate, dst_update, src_update);

  edge_kernel<<<(BB * EE) / 16, 256, 0, stream>>>(
      edge, ii, jj, Weg, beg, src_gate, dst_gate, dst_update,
      y_out, ssh, ssum);

  finalize_kernel<<<(int)((NNF + 255) / 256), 256, 0, stream>>>(
      node, src_update, ssh, ssum, x_out, (int)NNF);
}